// ComHG_attention_40604620816400
// MI455X (gfx1250) — compile-verified
//
#include <hip/hip_runtime.h>

#define N_NODES 50000
#define N_EDGES 1600000
#define DIN 128
#define DOUT 256
#define DA 32
#define SLOPE 0.2f

typedef __attribute__((ext_vector_type(2))) float v2f;
typedef __attribute__((ext_vector_type(8))) float v8f;

static __device__ __forceinline__ v8f wmma_f32_16x16x4(v2f a, v2f b, v8f c) {
  // D = A(16x4) * B(4x16) + C, full fp32
  return __builtin_amdgcn_wmma_f32_16x16x4_f32(false, a, false, b, (short)0, c,
                                               false, false);
}

// ---------------- Kernel 1: a_row = x@W_row, a_col = x@W_col (fp32 WMMA) ----
__global__ void proj_kernel(const float* __restrict__ x,
                            const float* __restrict__ W_row,
                            const float* __restrict__ W_col,
                            float* __restrict__ a_row,
                            float* __restrict__ a_col) {
  const int wave = (blockIdx.x * blockDim.x + threadIdx.x) >> 5;
  const int lane = threadIdx.x & 31;
  if (wave >= N_NODES / 16) return;          // wave-uniform guard (EXEC all-1)
  const int tile = wave;
  const int half = lane >> 4;                // 0: K=0,1   1: K=2,3
  const int m    = lane & 15;                // row (A) / col (B/C/D)
  const int node = tile * 16 + m;

  v8f acc0 = {}, acc1 = {}, acc2 = {}, acc3 = {};
  const float* xrow = x + (size_t)node * DIN + 2 * half;
  for (int k = 0; k < DIN; k += 4) {
    v2f a = *(const v2f*)(xrow + k);
    const int kk = k + 2 * half;
    v2f b0 = { W_row[kk * DA + m],      W_row[(kk + 1) * DA + m] };
    v2f b1 = { W_row[kk * DA + 16 + m], W_row[(kk + 1) * DA + 16 + m] };
    v2f b2 = { W_col[kk * DA + m],      W_col[(kk + 1) * DA + m] };
    v2f b3 = { W_col[kk * DA + 16 + m], W_col[(kk + 1) * DA + 16 + m] };
    acc0 = wmma_f32_16x16x4(a, b0, acc0);
    acc1 = wmma_f32_16x16x4(a, b1, acc1);
    acc2 = wmma_f32_16x16x4(a, b2, acc2);
    acc3 = wmma_f32_16x16x4(a, b3, acc3);
  }
#pragma unroll
  for (int v = 0; v < 8; ++v) {
    const int on = tile * 16 + v + half * 8;
    a_row[on * DA + m]      = acc0[v];
    a_row[on * DA + 16 + m] = acc1[v];
    a_col[on * DA + m]      = acc2[v];
    a_col[on * DA + 16 + m] = acc3[v];
  }
}

// ---------------- Kernel 2: raw edge scores + global max (uint-key) ---------
__global__ void edge_score_kernel(const float* __restrict__ a_row,
                                  const float* __restrict__ a_col,
                                  const int* __restrict__ row,
                                  const int* __restrict__ col,
                                  float* __restrict__ raw,
                                  unsigned* __restrict__ maxkey) {
  const int e = blockIdx.x * blockDim.x + threadIdx.x;
  if (e >= N_EDGES) return;
  const float4* ar = (const float4*)(a_row + (size_t)row[e] * DA);
  const float4* ac = (const float4*)(a_col + (size_t)col[e] * DA);
  float s = 0.f;
#pragma unroll
  for (int i = 0; i < DA / 4; ++i) {
    float4 u = ar[i], v = ac[i];
    s += u.x * v.x + u.y * v.y + u.z * v.z + u.w * v.w;
  }
  s *= 0.17677669529663687f;                 // rsqrt(32)
  s = (s >= 0.f) ? s : SLOPE * s;            // leaky_relu
  raw[e] = s;
  // order-preserving float->uint key; memset(0) is the identity element
  unsigned b = __float_as_uint(s);
  unsigned key = (b & 0x80000000u) ? ~b : (b | 0x80000000u);
#pragma unroll
  for (int off = 16; off > 0; off >>= 1) {
    unsigned o = __shfl_xor(key, off, 32);
    key = (o > key) ? o : key;
  }
  if ((threadIdx.x & 31) == 0) atomicMax(maxkey, key);
}

// ---------------- Kernel 3: exp(raw - gmax), deg = segment_sum --------------
__global__ void edge_exp_deg_kernel(const float* __restrict__ raw,
                                    const int* __restrict__ row,
                                    const unsigned* __restrict__ maxkey,
                                    float* __restrict__ atten,
                                    float* __restrict__ deg) {
  const int e = blockIdx.x * blockDim.x + threadIdx.x;
  if (e >= N_EDGES) return;
  const unsigned key = *maxkey;
  const unsigned b = (key & 0x80000000u) ? (key ^ 0x80000000u) : ~key;
  const float gmax = __uint_as_float(b);
  const float a = expf(raw[e] - gmax);
  atten[e] = a;
  atomicAdd(deg + row[e], a);
}

// ---------------- Kernel 4: SpMM scatter, one wave per edge -----------------
// Edge index is forced wave-uniform (readfirstlane) so all per-edge metadata
// (row/col/adj_val/atten/deg) lowers to the scalar-memory path, leaving the
// VMEM pipe for the x[col] float4 gathers and the packed f32 atomic stream.
__global__ void scatter_kernel(const float* __restrict__ x,
                               const int* __restrict__ row,
                               const int* __restrict__ col,
                               const float* __restrict__ adj_val,
                               const float* __restrict__ atten,
                               const float* __restrict__ deg,
                               float* __restrict__ feat) {
  const int wpb = blockDim.x >> 5;           // waves per block (8)
  const int e = __builtin_amdgcn_readfirstlane(blockIdx.x * wpb +
                                               (threadIdx.x >> 5));
  const int lane = threadIdx.x & 31;
  if (e >= N_EDGES) return;                  // wave-uniform guard
  const int r = row[e], c = col[e];
  const float val = (adj_val[e] + atten[e] / (deg[r] + 1e-15f)) * 0.5f;
  const float4 xv = *(const float4*)(x + (size_t)c * DIN + lane * 4);
  float* dst = feat + (size_t)r * DIN + lane * 4;
  atomicAdd(dst + 0, val * xv.x);
  atomicAdd(dst + 1, val * xv.y);
  atomicAdd(dst + 2, val * xv.z);
  atomicAdd(dst + 3, val * xv.w);
}

// ---------------- Kernel 5: out = feat @ W_x + b_x (fp32 WMMA) --------------
__global__ void out_gemm_kernel(const float* __restrict__ feat,
                                const float* __restrict__ W_x,
                                const float* __restrict__ b_x,
                                float* __restrict__ out) {
  const int wave = (blockIdx.x * blockDim.x + threadIdx.x) >> 5;
  const int lane = threadIdx.x & 31;
  const int total = (N_NODES / 16) * 4;      // row tiles x 4 col groups of 64
  if (wave >= total) return;                 // wave-uniform guard
  const int rt    = wave >> 2;
  const int cbase = (wave & 3) * 64;
  const int half  = lane >> 4;
  const int m     = lane & 15;
  const int node  = rt * 16 + m;

  v8f acc[4];
#pragma unroll
  for (int t = 0; t < 4; ++t) {
    const float bias = b_x[cbase + t * 16 + m];
#pragma unroll
    for (int v = 0; v < 8; ++v) acc[t][v] = bias;
  }
  const float* frow = feat + (size_t)node * DIN + 2 * half;
  for (int k = 0; k < DIN; k += 4) {
    v2f a = *(const v2f*)(frow + k);
    const int kk = k + 2 * half;
#pragma unroll
    for (int t = 0; t < 4; ++t) {
      v2f b = { W_x[kk * DOUT + cbase + t * 16 + m],
                W_x[(kk + 1) * DOUT + cbase + t * 16 + m] };
      acc[t] = wmma_f32_16x16x4(a, b, acc[t]);
    }
  }
#pragma unroll
  for (int t = 0; t < 4; ++t)
#pragma unroll
    for (int v = 0; v < 8; ++v)
      out[(size_t)(rt * 16 + v + half * 8) * DOUT + cbase + t * 16 + m] = acc[t][v];
}

extern "C" void kernel_launch(void* const* d_in, const int* in_sizes, int n_in,
                              void* d_out, int out_size, void* d_ws, size_t ws_size,
                              hipStream_t stream) {
  const float* x      = (const float*)d_in[0];
  const int*   row    = (const int*)d_in[1];
  const int*   col    = (const int*)d_in[2];
  const float* adjval = (const float*)d_in[3];
  const float* W_row  = (const float*)d_in[4];
  const float* W_col  = (const float*)d_in[5];
  const float* W_x    = (const float*)d_in[6];
  const float* b_x    = (const float*)d_in[7];
  float* out = (float*)d_out;

  char* ws = (char*)d_ws;
  size_t off = 0;
  float* a_row = (float*)(ws + off); off += (size_t)N_NODES * DA * 4;
  float* a_col = (float*)(ws + off); off += (size_t)N_NODES * DA * 4;
  float* raw   = (float*)(ws + off); off += (size_t)N_EDGES * 4;
  float* atten = (float*)(ws + off); off += (size_t)N_EDGES * 4;
  float* deg   = (float*)(ws + off); off += (size_t)N_NODES * 4;
  unsigned* maxkey = (unsigned*)(ws + off); off += 256;
  float* feat  = (float*)(ws + off); off += (size_t)N_NODES * DIN * 4;

  // accumulation buffers must be zeroed every call (deterministic replay)
  hipMemsetAsync(deg, 0, (size_t)N_NODES * 4, stream);
  hipMemsetAsync(maxkey, 0, 4, stream);
  hipMemsetAsync(feat, 0, (size_t)N_NODES * DIN * 4, stream);

  {
    const int waves = N_NODES / 16;                       // 3125
    const int blocks = (waves * 32 + 255) / 256;
    proj_kernel<<<blocks, 256, 0, stream>>>(x, W_row, W_col, a_row, a_col);
  }
  {
    const int blocks = (N_EDGES + 255) / 256;
    edge_score_kernel<<<blocks, 256, 0, stream>>>(a_row, a_col, row, col, raw, maxkey);
    edge_exp_deg_kernel<<<blocks, 256, 0, stream>>>(raw, row, maxkey, atten, deg);
  }
  {
    const int blocks = N_EDGES / 8;                       // 8 edges per block
    scatter_kernel<<<blocks, 256, 0, stream>>>(x, row, col, adjval, atten, deg, feat);
  }
  {
    const int waves = (N_NODES / 16) * 4;                 // 12500
    const int blocks = (waves * 32 + 255) / 256;
    out_gemm_kernel<<<blocks, 256, 0, stream>>>(feat, W_x, b_x, out);
  }
}